// SimpleFCTP_20796231647546
// MI455X (gfx1250) — compile-verified
//
#include <hip/hip_runtime.h>

// ---------------------------------------------------------------------------
// SimpleFCTP on MI455X (gfx1250, wave32):
//   out[j] = INV_SQRT3 * w[j] * sum(frac*frac)   -- memory-bound reduction.
// Pass 1: streaming b128 sum-of-squares -> per-block partials in d_ws.
// Pass 2: reduce partials, scale by w -> out[10].
// Wave reduction uses V_WMMA_F32_16X16X4_F32 (ones-matrix trick) + shfl_xor.
// ---------------------------------------------------------------------------

typedef __attribute__((ext_vector_type(2))) float v2f;
typedef __attribute__((ext_vector_type(4))) float fvec4;
typedef __attribute__((ext_vector_type(8))) float v8f;

#define INV_SQRT3f 0.57735026918962576f

// Reduce 32 per-lane partials to a wave total (result in every lane).
// Stage 1: WMMA with A = ones(16x4), B = {acc, 0}:
//   D[m][n] = acc[n] + acc[n+16] for n in 0..15, replicated over all m,
//   so c[0] holds the 16 pairwise sums (duplicated in lanes 16..31).
// Stage 2: butterfly over the 16 values.
// Requires EXEC == all ones (call only from fully-converged code).
__device__ __forceinline__ float wave_reduce_all(float acc) {
    v2f a; a.x = 1.0f; a.y = 1.0f;          // A matrix: all ones
    v2f b; b.x = acc;  b.y = 0.0f;          // B matrix: partials in one VGPR row-pair
    v8f c = {};
    c = __builtin_amdgcn_wmma_f32_16x16x4_f32(
        /*neg_a=*/false, a, /*neg_b=*/false, b,
        /*c_mod=*/(short)0, c, /*reuse_a=*/false, /*reuse_b=*/false);
    float s = c[0];
    s += __shfl_xor(s, 1, 32);
    s += __shfl_xor(s, 2, 32);
    s += __shfl_xor(s, 4, 32);
    s += __shfl_xor(s, 8, 32);
    return s;
}

__global__ void __launch_bounds__(256)
sumsq_partials_kernel(const float* __restrict__ x, long nvec4, int ntail,
                      float* __restrict__ partials) {
    const fvec4* __restrict__ xv = (const fvec4*)x;
    const long stride = (long)gridDim.x * (long)blockDim.x;
    const long base   = (long)blockIdx.x * (long)blockDim.x + threadIdx.x;

    // 4 independent accumulators, 4 b128 nontemporal loads in flight.
    float a0 = 0.f, a1 = 0.f, a2 = 0.f, a3 = 0.f;
    long i = base;
    for (; i + 3 * stride < nvec4; i += 4 * stride) {
        fvec4 v0 = __builtin_nontemporal_load(&xv[i]);
        fvec4 v1 = __builtin_nontemporal_load(&xv[i + stride]);
        fvec4 v2 = __builtin_nontemporal_load(&xv[i + 2 * stride]);
        fvec4 v3 = __builtin_nontemporal_load(&xv[i + 3 * stride]);
        a0 = fmaf(v0.x, v0.x, a0); a0 = fmaf(v0.y, v0.y, a0);
        a0 = fmaf(v0.z, v0.z, a0); a0 = fmaf(v0.w, v0.w, a0);
        a1 = fmaf(v1.x, v1.x, a1); a1 = fmaf(v1.y, v1.y, a1);
        a1 = fmaf(v1.z, v1.z, a1); a1 = fmaf(v1.w, v1.w, a1);
        a2 = fmaf(v2.x, v2.x, a2); a2 = fmaf(v2.y, v2.y, a2);
        a2 = fmaf(v2.z, v2.z, a2); a2 = fmaf(v2.w, v2.w, a2);
        a3 = fmaf(v3.x, v3.x, a3); a3 = fmaf(v3.y, v3.y, a3);
        a3 = fmaf(v3.z, v3.z, a3); a3 = fmaf(v3.w, v3.w, a3);
    }
    for (; i < nvec4; i += stride) {
        fvec4 v = __builtin_nontemporal_load(&xv[i]);
        a0 = fmaf(v.x, v.x, a0); a0 = fmaf(v.y, v.y, a0);
        a0 = fmaf(v.z, v.z, a0); a0 = fmaf(v.w, v.w, a0);
    }
    float acc = (a0 + a1) + (a2 + a3);

    // Tail floats (size % 4 <= 3): handled by the first few global threads.
    if (base < (long)ntail) {
        float t = x[nvec4 * 4 + base];
        acc = fmaf(t, t, acc);
    }
    // Fully converged here: EXEC all ones for WMMA.
    float wsum = wave_reduce_all(acc);

    __shared__ float lds[8];
    const int lane = threadIdx.x & 31;
    const int wid  = threadIdx.x >> 5;
    if (lane == 0) lds[wid] = wsum;
    __syncthreads();
    if (wid == 0) {
        float v = (lane < 8) ? lds[lane] : 0.f;
        v += __shfl_xor(v, 1, 32);
        v += __shfl_xor(v, 2, 32);
        v += __shfl_xor(v, 4, 32);
        if (lane == 0) partials[blockIdx.x] = v;
    }
}

__global__ void __launch_bounds__(256)
finalize_kernel(const float* __restrict__ partials, int nparts,
                const float* __restrict__ w, float* __restrict__ out, int outn) {
    float acc = 0.f;
    for (int i = threadIdx.x; i < nparts; i += 256) acc += partials[i];

    // Fully converged here: EXEC all ones for WMMA.
    float wsum = wave_reduce_all(acc);

    __shared__ float lds[8];
    __shared__ float Ssh;
    const int lane = threadIdx.x & 31;
    const int wid  = threadIdx.x >> 5;
    if (lane == 0) lds[wid] = wsum;
    __syncthreads();
    if (threadIdx.x == 0) {
        float t = 0.f;
        #pragma unroll
        for (int i = 0; i < 8; ++i) t += lds[i];
        Ssh = t * INV_SQRT3f;
    }
    __syncthreads();
    if ((int)threadIdx.x < outn) out[threadIdx.x] = Ssh * w[threadIdx.x];
}

extern "C" void kernel_launch(void* const* d_in, const int* in_sizes, int n_in,
                              void* d_out, int out_size, void* d_ws, size_t ws_size,
                              hipStream_t stream) {
    const float* frac = (const float*)d_in[0];   // [N,3] flat = 3N floats
    const float* w    = (const float*)d_in[1];   // [10]
    float*       out  = (float*)d_out;           // [10]
    float*       partials = (float*)d_ws;

    const long total = (long)in_sizes[0];        // 3N floats
    const long nvec4 = total >> 2;
    const int  ntail = (int)(total & 3);

    // 3072 blocks x 256 threads: for N=16.7M this is exactly 16 b128 loads
    // per thread. Clamp to workspace capacity (1 float per block).
    int nblocks = 3072;
    const long maxblocks = (long)(ws_size / sizeof(float));
    if ((long)nblocks > maxblocks) nblocks = (int)(maxblocks > 0 ? maxblocks : 1);

    sumsq_partials_kernel<<<nblocks, 256, 0, stream>>>(frac, nvec4, ntail, partials);
    finalize_kernel<<<1, 256, 0, stream>>>(partials, nblocks, w, out, out_size);
}